// StochasticPool2d_74285754351770
// MI455X (gfx1250) — compile-verified
//
#include <hip/hip_runtime.h>
#include <cstdint>

// StochasticPool2d (eval): out[b,c,oh,ow] = sum(x^2 over 2x2 win) / sum(x over 2x2 win)
// x: (16,128,256,256) f32 -> out: (16,128,128,128) f32
//
// Pure streaming / HBM-bandwidth-bound (AI ~ 0.55 flop/byte, floor ~28us @ 23.3TB/s).
// Strategy: one wave32 per output row. The wave's two source rows are 512
// contiguous floats; DMA them into LDS with CDNA5 async global->LDS b128 loads
// (ASYNCcnt), wait with s_wait_asynccnt, compute from LDS (ds_load_b128), and
// write the output row with non-temporal coalesced stores.

typedef __attribute__((ext_vector_type(4))) float v4f;
typedef __attribute__((ext_vector_type(2))) float v2f;

#define WAVES_PER_BLOCK 8u
#define ROW_FLOATS 512u              // two input rows of 256 floats, contiguous
#define H_OUT 128u
#define W_OUT 128u

__global__ __launch_bounds__(256) void stochpool2x2_kernel(
    const float* __restrict__ in, float* __restrict__ out) {
  // 8 waves x 2KiB = 16 KiB LDS per block
  __shared__ alignas(16) float sbuf[WAVES_PER_BLOCK * ROW_FLOATS];

  const unsigned tid  = threadIdx.x;
  const unsigned w    = tid >> 5;    // wave in block
  const unsigned lane = tid & 31u;

  // Flat output-row index R in [0, B*C*H_out) = [0, 262144)
  const unsigned R  = blockIdx.x * WAVES_PER_BLOCK + w;
  const unsigned oh = R & (H_OUT - 1u);  // output row within image
  const unsigned bc = R >> 7;            // fused batch*channel

  // Input rows 2*oh and 2*oh+1 of image bc: 512 contiguous floats.
  // Byte offset from `in`: bc*256*256*4 + oh*2*256*4  (max ~536MB, fits u32)
  const unsigned gvoff = (bc << 18) + (oh << 11) + lane * 16u;

  // 32-bit LDS byte address for this wave's buffer (low 32 bits of the flat
  // shared-aperture pointer == offset within the workgroup's LDS allocation).
  const unsigned lbase = (unsigned)(uintptr_t)(&sbuf[w * ROW_FLOATS]) + lane * 16u;

  // Async DMA: 4 x (32 lanes x 16B) = 2048B -> LDS, tracked by ASYNCcnt.
#pragma unroll
  for (unsigned k = 0; k < 4; ++k) {
    asm volatile("global_load_async_to_lds_b128 %0, %1, %2"
                 :
                 : "v"(lbase + k * 512u), "v"(gvoff + k * 512u), "s"(in)
                 : "memory");
  }
  asm volatile("s_wait_asynccnt 0" ::: "memory");
  // Each wave reads only the LDS it wrote itself -> no barrier needed.

  const float* row = &sbuf[w * ROW_FLOATS];       // floats 0..255: row 2*oh
                                                  // floats 256..511: row 2*oh+1
  float* orow = out + (size_t)R * W_OUT;

#pragma unroll
  for (unsigned c = 0; c < 2; ++c) {
    const unsigned fo = c * 128u + 4u * lane;     // input col offset (floats)
    const v4f a = *(const v4f*)(row + fo);        // top row, 4 cols
    const v4f b = *(const v4f*)(row + 256u + fo); // bottom row, 4 cols

    // Two 2x2 windows: cols (fo,fo+1) -> out col c*64+2*lane,
    //                  cols (fo+2,fo+3) -> out col c*64+2*lane+1
    const float s1x = (a.x + a.y) + (b.x + b.y);
    const float s2x = (a.x * a.x + a.y * a.y) + (b.x * b.x + b.y * b.y);
    const float s1y = (a.z + a.w) + (b.z + b.w);
    const float s2y = (a.z * a.z + a.w * a.w) + (b.z * b.z + b.w * b.w);

    v2f r;
    r.x = s2x / s1x;
    r.y = s2y / s1y;

    // Output is write-once, never re-read -> non-temporal store.
    __builtin_nontemporal_store(r, (v2f*)(orow + c * 64u + 2u * lane));
  }
}

extern "C" void kernel_launch(void* const* d_in, const int* in_sizes, int n_in,
                              void* d_out, int out_size, void* d_ws, size_t ws_size,
                              hipStream_t stream) {
  const float* x = (const float*)d_in[0];
  float* out = (float*)d_out;

  // B*C*H_out = 16*128*128 = 262144 output rows; 8 rows (waves) per block.
  const unsigned rows = 16u * 128u * 128u;
  dim3 grid(rows / WAVES_PER_BLOCK);   // 32768
  dim3 block(256);
  stochpool2x2_kernel<<<grid, block, 0, stream>>>(x, out);
}